// CombinedLoss_80891414053477
// MI455X (gfx1250) — compile-verified
//
#include <hip/hip_runtime.h>

typedef __attribute__((ext_vector_type(2))) float v2f;
typedef __attribute__((ext_vector_type(8))) float v8f;

#define WS   11
#define PAD  5
#define PH   26          // patch rows (16 + 2*5)
#define PW   28          // patch cols padded to 7 K-chunks of 4
#define TILE 16
#define NB   256         // reduction buckets
#define HTS  (TILE * PW) // transposed H buffer per quantity: [16 cols][28 rows]

static __device__ __forceinline__ v2f mkv2(float a, float b) {
    v2f r; r[0] = a; r[1] = b; return r;
}
static __device__ __forceinline__ v8f zero8() {
    v8f r;
#pragma unroll
    for (int i = 0; i < 8; ++i) r[i] = 0.0f;
    return r;
}

// D(16x16,f32) = A(16x4,f32) * B(4x16,f32) + C  — full fp32 matrix pipe on CDNA5
static __device__ __forceinline__ v8f wmma4(v2f a, v2f b, v8f c) {
    return __builtin_amdgcn_wmma_f32_16x16x4_f32(
        /*neg_a=*/false, a, /*neg_b=*/false, b,
        /*c_mod=*/(short)0, c, /*reuse_a=*/false, /*reuse_b=*/false);
}

__global__ void __launch_bounds__(32)
ssim_tile_kernel(const float* __restrict__ x, const float* __restrict__ y,
                 const float* __restrict__ win, float* __restrict__ buckets,
                 int H, int W) {
    __shared__ __align__(16) float xs[PH * PW];
    __shared__ __align__(16) float ys[PH * PW];
    __shared__ __align__(16) float ht[5][HTS];

    const int lane = threadIdx.x;   // 0..31, wave32
    const int half = lane >> 4;     // lane group (K split)
    const int mn   = lane & 15;     // M (A rows) / N (B cols)

    const int r0 = blockIdx.y * TILE;
    const int c0 = blockIdx.x * TILE;
    const long imgOff = (long)blockIdx.z * H * W;
    const float* xb = x + imgOff;
    const float* yb = y + imgOff;

    // ---- band-weight constants in VGPRs.
    // window = g1*g1^T, so g1[d] = window[d*11+5] / sqrt(window[5*11+5]).
    // Lane holds W[k][n] = g1[k-n] with k = 4*kc + 2*half + j.
    // Under the f32 WMMA layouts this same register set is B for the
    // horizontal pass and A (the band matrix G[m][k]=g1[k-m]) for the vertical.
    const float invc = 1.0f / sqrtf(win[PAD * WS + PAD]);
    v2f wb[7];
#pragma unroll
    for (int kc = 0; kc < 7; ++kc) {
#pragma unroll
        for (int j = 0; j < 2; ++j) {
            int k = 4 * kc + 2 * half + j;
            int d = k - mn;
            float w = 0.0f;
            if (d >= 0 && d <= 10) w = win[d * WS + PAD] * invc;
            wb[kc][j] = w;
        }
    }

    // ---- load zero-padded 26x28 patch into LDS; L1 term on the 16x16 interior
    float l1 = 0.0f;
    for (int i = lane; i < PH * PW; i += 32) {
        int pr = i / PW;
        int pc = i - pr * PW;
        int gr = r0 + pr - PAD;
        int gc = c0 + pc - PAD;
        float vx = 0.0f, vy = 0.0f;
        if (pc < PH && gr >= 0 && gr < H && gc >= 0 && gc < W) {
            int idx = gr * W + gc;
            vx = xb[idx];
            vy = yb[idx];
        }
        xs[i] = vx;
        ys[i] = vy;
        if (pr >= PAD && pr < PAD + TILE && pc >= PAD && pc < PAD + TILE)
            l1 += fabsf(vx - vy);
    }

    // zero Ht K-pad rows 26..27 so the kc=6 chunk multiplies 0*0, never 0*garbage
    if (lane < TILE) {
#pragma unroll
        for (int q = 0; q < 5; ++q)
            *(v2f*)&ht[q][lane * PW + PH] = mkv2(0.0f, 0.0f);
    }

    // ---- pass 1: horizontal blur of x, y, x^2, y^2, xy.
    // Need 26 blurred rows -> two overlapping 16-row groups (rb = 0 and 10).
    // Per chunk: A = patch rows (ds_load_b64), B = band weights (wb).
#pragma unroll
    for (int grp = 0; grp < 2; ++grp) {
        const int rb = grp * (PH - TILE); // 0, 10
        v8f acc[5];
#pragma unroll
        for (int q = 0; q < 5; ++q) acc[q] = zero8();
#pragma unroll
        for (int kc = 0; kc < 7; ++kc) {
            const int k0 = 4 * kc + 2 * half;
            const int arow = (rb + mn) * PW + k0;
            v2f ax = *(const v2f*)&xs[arow];
            v2f ay = *(const v2f*)&ys[arow];
            acc[0] = wmma4(ax,      wb[kc], acc[0]);
            acc[1] = wmma4(ay,      wb[kc], acc[1]);
            acc[2] = wmma4(ax * ax, wb[kc], acc[2]);
            acc[3] = wmma4(ay * ay, wb[kc], acc[3]);
            acc[4] = wmma4(ax * ay, wb[kc], acc[4]);
        }
        // store transposed: ht[q][col n][row m], D VGPR v holds row v + 8*half
        const int base = mn * PW + rb + 8 * half;
#pragma unroll
        for (int q = 0; q < 5; ++q) {
#pragma unroll
            for (int v = 0; v < 8; v += 2)
                *(v2f*)&ht[q][base + v] = mkv2(acc[q][v], acc[q][v + 1]);
        }
    }

    // ---- pass 2: vertical blur. A = band weights (wb), B = Ht columns.
    v8f o[5];
#pragma unroll
    for (int q = 0; q < 5; ++q) o[q] = zero8();
#pragma unroll
    for (int kc = 0; kc < 7; ++kc) {
        const int k0 = mn * PW + 4 * kc + 2 * half;
#pragma unroll
        for (int q = 0; q < 5; ++q) {
            v2f b = *(const v2f*)&ht[q][k0];
            o[q] = wmma4(wb[kc], b, o[q]);
        }
    }

    // ---- SSIM map on the 16x16 tile (8 elements per lane) + tile sum
    const float C1 = 1.0e-4f;  // (0.01*VAL_RANGE)^2
    const float C2 = 9.0e-4f;  // (0.03*VAL_RANGE)^2
    v8f mu1 = o[0], mu2 = o[1];
    v8f mu1s = mu1 * mu1;
    v8f mu2s = mu2 * mu2;
    v8f mu12 = mu1 * mu2;
    v8f s1  = o[2] - mu1s;
    v8f s2  = o[3] - mu2s;
    v8f s12 = o[4] - mu12;
    v8f num = (mu12 + mu12 + C1) * (s12 + s12 + C2);
    v8f den = (mu1s + mu2s + C1) * (s1 + s2 + C2);
    v8f sm  = num / den;
    float s = sm[0] + sm[1] + sm[2] + sm[3] + sm[4] + sm[5] + sm[6] + sm[7];

    // ---- wave32 reduction, one atomic pair per tile into scattered buckets
#pragma unroll
    for (int off = 16; off > 0; off >>= 1) {
        s  += __shfl_xor(s,  off, 32);
        l1 += __shfl_xor(l1, off, 32);
    }
    if (lane == 0) {
        int b = (blockIdx.x + 32 * blockIdx.y + 1024 * blockIdx.z) & (NB - 1);
        atomicAdd(&buckets[b], s);
        atomicAdd(&buckets[NB + b], l1);
    }
}

__global__ void init_buckets(float* __restrict__ b) {
    int i = threadIdx.x;
    if (i < 2 * NB) b[i] = 0.0f;
}

__global__ void __launch_bounds__(32)
finish_kernel(const float* __restrict__ b, float* __restrict__ out, float invN) {
    int lane = threadIdx.x;
    float s = 0.0f, l = 0.0f;
    for (int i = lane; i < NB; i += 32) {
        s += b[i];
        l += b[NB + i];
    }
#pragma unroll
    for (int off = 16; off > 0; off >>= 1) {
        s += __shfl_xor(s, off, 32);
        l += __shfl_xor(l, off, 32);
    }
    if (lane == 0) {
        float ssim_mean = s * invN;
        float l1_mean   = l * invN;
        // ALPHA*(1-ssim) + (1-ALPHA)*l1, ALPHA=0.5
        out[0] = 0.5f * (1.0f - ssim_mean) + 0.5f * l1_mean;
    }
}

extern "C" void kernel_launch(void* const* d_in, const int* in_sizes, int n_in,
                              void* d_out, int out_size, void* d_ws, size_t ws_size,
                              hipStream_t stream) {
    (void)n_in; (void)out_size; (void)ws_size;
    const float* x   = (const float*)d_in[0]; // output (B,1,512,512) fp32
    const float* y   = (const float*)d_in[1]; // target
    const float* win = (const float*)d_in[2]; // (1,1,11,11) gaussian outer product
    float* out = (float*)d_out;
    float* buckets = (float*)d_ws;            // 2*NB floats

    const int H = 512, W = 512;
    const int B = in_sizes[0] / (H * W);

    hipLaunchKernelGGL(init_buckets, dim3(1), dim3(512), 0, stream, buckets);

    dim3 grid(W / TILE, H / TILE, B);
    hipLaunchKernelGGL(ssim_tile_kernel, grid, dim3(32), 0, stream,
                       x, y, win, buckets, H, W);

    const float invN = 1.0f / (float)((long)B * H * W);
    hipLaunchKernelGGL(finish_kernel, dim3(1), dim3(32), 0, stream,
                       buckets, out, invN);
}